// LRMRec_Encoder_4355096839070
// MI455X (gfx1250) — compile-verified
//
#include <hip/hip_runtime.h>
#include <hip/hip_bf16.h>

typedef float v2f __attribute__((ext_vector_type(2)));
typedef float v8f __attribute__((ext_vector_type(8)));

// ---------------------------------------------------------------------------
// Y[N,64] = X[N,64] @ W[64,64] using V_WMMA_F32_16X16X4_F32 (wave32).
//
// Block = 256 threads = 8 waves; each wave owns a 16-row x 64-col output
// strip (4 f32 16x16 accumulators).  W is staged once per block into LDS in a
// pair-interleaved, bank-padded layout:
//     wlds[(k>>1)*160 + col*2 + (k&1)] = W[k][col]
// so each B fragment (W[ks][col], W[ks+1][col]) is a single aligned
// ds_load_b64, and the 160-dword row stride (160 mod 64 == 32) puts the two
// half-waves on disjoint bank sets.  All 16 A fragments are preloaded as one
// global-load clause, so the K-loop is ds_load + wmma only.
//
// ISA layouts (CDNA5 ISA §7.12.2):
//   A 16x4 f32 : lanes 0-15 -> M=lane, VGPRs=K+0,K+1 ; lanes 16-31 -> K+2,K+3
//   B 4x16 f32 : lanes 0-15 -> N=lane, VGPRs=K+0,K+1 ; lanes 16-31 -> K+2,K+3
//   C/D 16x16  : VGPR r: lanes 0-15 -> (M=r, N=lane); lanes 16-31 -> (M=r+8)
// ---------------------------------------------------------------------------
#define LDS_WSTRIDE 160  // dwords per k-pair row (128 data + 32 pad)

__global__ __launch_bounds__(256) void node_gemm64(
    const float* __restrict__ X, const float* __restrict__ W,
    float* __restrict__ Y, int nrows) {
  __shared__ float wlds[32 * LDS_WSTRIDE];  // 20 KB

  const int tid = threadIdx.x;
  // Cooperative staging of W (4096 floats) into the interleaved layout.
#pragma unroll
  for (int i = tid; i < 4096; i += 256) {
    const int k = i >> 6, col = i & 63;
    wlds[(k >> 1) * LDS_WSTRIDE + col * 2 + (k & 1)] = W[i];
  }
  __syncthreads();

  const int wave = tid >> 5;
  const int lane = tid & 31;
  const int half = lane >> 4;  // 0: lanes 0-15, 1: lanes 16-31
  const int l16  = lane & 15;

  const int m0 = (blockIdx.x * 8 + wave) * 16;
  int arow = m0 + l16;
  if (arow >= nrows) arow = nrows - 1;  // clamp loads; EXEC stays all-1s

  // Preload all A fragments for this lane (one clause, 32 VGPRs).
  const float* __restrict__ ap = X + (size_t)arow * 64 + half * 2;
  v2f afrag[16];
#pragma unroll
  for (int kk = 0; kk < 16; ++kk) {
    afrag[kk] = *reinterpret_cast<const v2f*>(ap + kk * 4);
  }

  v8f acc[4] = {};
#pragma unroll
  for (int kk = 0; kk < 16; ++kk) {
    const float* __restrict__ brow = &wlds[(kk * 2 + half) * LDS_WSTRIDE];
#pragma unroll
    for (int t = 0; t < 4; ++t) {
      const v2f b = *reinterpret_cast<const v2f*>(brow + (t * 16 + l16) * 2);
      acc[t] = __builtin_amdgcn_wmma_f32_16x16x4_f32(
          /*neg_a=*/false, afrag[kk], /*neg_b=*/false, b,
          /*c_mod=*/(short)0, acc[t], /*reuse_a=*/false, /*reuse_b=*/false);
    }
  }

  const int orow = m0 + half * 8;
#pragma unroll
  for (int r = 0; r < 8; ++r) {
    if (orow + r < nrows) {
#pragma unroll
      for (int t = 0; t < 4; ++t) {
        Y[(size_t)(orow + r) * 64 + t * 16 + l16] = acc[t][r];
      }
    }
  }
}

// ---------------------------------------------------------------------------
// embeds0 = concat(user_emb, item_emb); also seeds the output accumulator.
// ---------------------------------------------------------------------------
__global__ void init_concat(const float4* __restrict__ u,
                            const float4* __restrict__ it,
                            float4* __restrict__ cur, float4* __restrict__ acc,
                            int uQuads, int totQuads) {
  int i = blockIdx.x * blockDim.x + threadIdx.x;
  if (i >= totQuads) return;
  float4 v = (i < uQuads) ? u[i] : it[i - uQuads];
  cur[i] = v;
  acc[i] = v;
}

__global__ void accum_add(float4* __restrict__ acc,
                          const float4* __restrict__ x, int nQuads) {
  int i = blockIdx.x * blockDim.x + threadIdx.x;
  if (i >= nQuads) return;
  float4 a = acc[i], b = x[i];
  a.x += b.x; a.y += b.y; a.z += b.z; a.w += b.w;
  acc[i] = a;
}

// ---------------------------------------------------------------------------
// Pass 1: attention logits -> exp -> scatter-sum softmax denominator.
// One thread per (edge, head): 4 consecutive lanes cover one contiguous
// 256 B Q/K row (coalesced); rows hit the 192 MB L2 (Q+K+V = 115 MB).
// ---------------------------------------------------------------------------
__global__ void edge_att(const int* __restrict__ rows,
                         const int* __restrict__ cols,
                         const float* __restrict__ Q,
                         const float* __restrict__ K,
                         float* __restrict__ att,      // [E,4]
                         float* __restrict__ attNorm,  // [N,4] pre-zeroed
                         int E) {
  int idx = blockIdx.x * blockDim.x + threadIdx.x;
  if (idx >= E * 4) return;
  const int e = idx >> 2, h = idx & 3;
  const int r = rows[e], c = cols[e];
  const float4* q4 = reinterpret_cast<const float4*>(Q + (size_t)r * 64 + h * 16);
  const float4* k4 = reinterpret_cast<const float4*>(K + (size_t)c * 64 + h * 16);
  float s = 0.f;
#pragma unroll
  for (int i = 0; i < 4; ++i) {
    float4 qv = q4[i];
    float4 kv = k4[i];
    s += qv.x * kv.x + qv.y * kv.y + qv.z * kv.z + qv.w * kv.w;
  }
  s = fminf(10.f, fmaxf(-10.f, s));
  float ea = expf(s);
  att[idx] = ea;
  atomicAdd(&attNorm[(size_t)r * 4 + h], ea);
}

// ---------------------------------------------------------------------------
// Pass 2: normalized attention * V scattered into destination rows.
// ---------------------------------------------------------------------------
__global__ void edge_agg(const int* __restrict__ rows,
                         const int* __restrict__ cols,
                         const float* __restrict__ V,
                         const float* __restrict__ att,
                         const float* __restrict__ attNorm,
                         float* __restrict__ Y,  // [N,64] pre-zeroed
                         int E) {
  int idx = blockIdx.x * blockDim.x + threadIdx.x;
  if (idx >= E * 4) return;
  const int e = idx >> 2, h = idx & 3;
  const int r = rows[e], c = cols[e];
  const float w = att[idx] / (attNorm[(size_t)r * 4 + h] + 1e-8f);
  const float4* v4 = reinterpret_cast<const float4*>(V + (size_t)c * 64 + h * 16);
  float* y = Y + (size_t)r * 64 + h * 16;
#pragma unroll
  for (int i = 0; i < 4; ++i) {
    float4 v = v4[i];
    atomicAdd(y + i * 4 + 0, w * v.x);
    atomicAdd(y + i * 4 + 1, w * v.y);
    atomicAdd(y + i * 4 + 2, w * v.z);
    atomicAdd(y + i * 4 + 3, w * v.w);
  }
}

// ---------------------------------------------------------------------------
// GCN layer: Y[rows[e]] += vals[e] * X[cols[e]].
// ---------------------------------------------------------------------------
__global__ void spmm_edge(const int* __restrict__ rows,
                          const int* __restrict__ cols,
                          const float* __restrict__ vals,
                          const float* __restrict__ X,
                          float* __restrict__ Y,  // pre-zeroed
                          int E) {
  int idx = blockIdx.x * blockDim.x + threadIdx.x;
  if (idx >= E * 4) return;
  const int e = idx >> 2, q = idx & 3;
  const int r = rows[e], c = cols[e];
  const float a = vals[e];
  const float4* x4 = reinterpret_cast<const float4*>(X + (size_t)c * 64 + q * 16);
  float* y = Y + (size_t)r * 64 + q * 16;
#pragma unroll
  for (int i = 0; i < 4; ++i) {
    float4 v = x4[i];
    atomicAdd(y + i * 4 + 0, a * v.x);
    atomicAdd(y + i * 4 + 1, a * v.y);
    atomicAdd(y + i * 4 + 2, a * v.z);
    atomicAdd(y + i * 4 + 3, a * v.w);
  }
}

// ---------------------------------------------------------------------------
extern "C" void kernel_launch(void* const* d_in, const int* in_sizes, int n_in,
                              void* d_out, int out_size, void* d_ws,
                              size_t ws_size, hipStream_t stream) {
  const int*   rows     = (const int*)d_in[0];
  const int*   cols     = (const int*)d_in[1];
  const float* vals     = (const float*)d_in[2];
  const float* user_emb = (const float*)d_in[3];
  const float* item_emb = (const float*)d_in[4];
  const float* Wq       = (const float*)d_in[5];
  const float* Wk       = (const float*)d_in[6];
  const float* Wv       = (const float*)d_in[7];

  const int E  = in_sizes[0];
  const int nU = in_sizes[3] / 64;
  const int nI = in_sizes[4] / 64;
  const int N  = nU + nI;

  float* out = (float*)d_out;

  // Workspace layout (floats): cur | nxt | Q | K (V aliases Q) | att | attNorm
  const size_t nd = (size_t)N * 64;
  float* cur  = (float*)d_ws;
  float* nxt  = cur + nd;
  float* Qb   = nxt + nd;
  float* Kb   = Qb + nd;
  float* att  = Kb + nd;
  float* attN = att + (size_t)E * 4;

  const int threads = 256;
  const int nQuads     = (int)(nd / 4);
  const int ndBlocks   = (nQuads + threads - 1) / threads;
  const int rowTiles   = (N + 15) / 16;
  const int gemmBlocks = (rowTiles + 7) / 8;  // 8 wave-tiles per block
  const int e4Blocks   = (E * 4 + threads - 1) / threads;

  // embeds0 = concat(user, item); out accumulator seeded with embeds0
  init_concat<<<ndBlocks, threads, 0, stream>>>(
      (const float4*)user_emb, (const float4*)item_emb, (float4*)cur,
      (float4*)out, nU * 16, nQuads);

  // ---- 2 graph-transformer layers ----
  for (int l = 0; l < 2; ++l) {
    const float* wq = Wq + (size_t)l * 64 * 64;
    const float* wk = Wk + (size_t)l * 64 * 64;
    const float* wv = Wv + (size_t)l * 64 * 64;

    node_gemm64<<<gemmBlocks, 256, 0, stream>>>(cur, wq, Qb, N);
    node_gemm64<<<gemmBlocks, 256, 0, stream>>>(cur, wk, Kb, N);
    hipMemsetAsync(attN, 0, (size_t)N * 4 * sizeof(float), stream);
    edge_att<<<e4Blocks, threads, 0, stream>>>(rows, cols, Qb, Kb, att, attN, E);

    node_gemm64<<<gemmBlocks, 256, 0, stream>>>(cur, wv, Qb, N);  // V -> Qb
    hipMemsetAsync(nxt, 0, nd * sizeof(float), stream);
    edge_agg<<<e4Blocks, threads, 0, stream>>>(rows, cols, Qb, att, attN, nxt, E);

    accum_add<<<ndBlocks, threads, 0, stream>>>((float4*)out,
                                                (const float4*)nxt, nQuads);
    float* t = cur; cur = nxt; nxt = t;
  }

  // ---- 2 GCN (spmm) layers ----
  for (int l = 0; l < 2; ++l) {
    hipMemsetAsync(nxt, 0, nd * sizeof(float), stream);
    spmm_edge<<<e4Blocks, threads, 0, stream>>>(rows, cols, vals, cur, nxt, E);
    accum_add<<<ndBlocks, threads, 0, stream>>>((float4*)out,
                                                (const float4*)nxt, nQuads);
    float* t = cur; cur = nxt; nxt = t;
  }
}